// IMUToTrajectoryNet_77094662963457
// MI455X (gfx1250) — compile-verified
//
#include <hip/hip_runtime.h>

// GRU (B=128, T=4096, I=6, H=32) + FC(32->2), fully fused, f32 WMMA on gfx1250.
//
// Parallelization: grid = 8 blocks (16 batch rows each), 2 waves/block.
//   wave w owns h columns [16w, 16w+16): its r/z/n 16x16 output tiles.
//   wave 0 additionally computes the FC tile from h_{t-1} (one-step pipeline
//   shift so FC reuses the same WMMA A-operands as the recurrence).
// Per step per wave: 6 gx WMMA (K=8 padded) + 24 gh WMMA (K=32) [+8 FC WMMA
// on wave 0], all V_WMMA_F32_16X16X4_F32 chains accumulating in registers.
// h is exchanged between the two waves through LDS in h^T layout
// (addr = (k*16 + m)*4): the D-layout producer writes contiguous 8-float
// runs, the A-layout consumer reads 2 dwords per K-slice.

typedef float v2f __attribute__((ext_vector_type(2)));
typedef float v8f __attribute__((ext_vector_type(8)));

#define B_ 128
#define T_ 4096
#define I_ 6
#define H_ 32
#define O_ 2

__device__ __forceinline__ v8f wmma4(v2f a, v2f b, v8f c) {
    // D = A(16x4 f32) * B(4x16 f32) + C(16x16 f32)
    return __builtin_amdgcn_wmma_f32_16x16x4_f32(
        /*neg_a=*/false, a, /*neg_b=*/false, b,
        /*c_mod=*/(short)0, c, /*reuse_a=*/false, /*reuse_b=*/false);
}

__device__ __forceinline__ float fsigmoid(float a) {
    // 1 / (1 + exp(-a)) via v_exp_f32 (exp2) + v_rcp_f32
    return __builtin_amdgcn_rcpf(1.0f + __builtin_amdgcn_exp2f(-1.44269504088896f * a));
}
__device__ __forceinline__ float ftanh(float a) {
    // tanh(a) = 2*sigmoid(2a) - 1
    return 2.0f * __builtin_amdgcn_rcpf(1.0f + __builtin_amdgcn_exp2f(-2.88539008177793f * a)) - 1.0f;
}

__global__ __launch_bounds__(64, 1) void gru_fused_kernel(
    const float* __restrict__ x,        // (B, T, I)
    const int*   __restrict__ lengths,  // (B,)
    const float* __restrict__ w_ih,     // (3H, I)
    const float* __restrict__ w_hh,     // (3H, H)
    const float* __restrict__ b_ih,     // (3H,)
    const float* __restrict__ b_hh,     // (3H,)
    const float* __restrict__ fc_w,     // (O, H)
    const float* __restrict__ fc_b,     // (O,)
    float*       __restrict__ out)      // (B, T, O)
{
    __shared__ float ldsh[H_ * 16];     // h^T layout: ldsh[k*16 + m], k=h-index, m=batch row

    const int tid  = threadIdx.x;
    const int wv   = tid >> 5;          // wave id: 0 or 1
    const int lane = tid & 31;
    const int lh   = lane >> 4;         // lane-half (selects K/M sub-block)
    const int ln   = lane & 15;         // column (B/C/D) or row (A) index
    const int b0   = blockIdx.x * 16;

    // h0 = 0
    for (int j = tid; j < H_ * 16; j += 64) ldsh[j] = 0.0f;
    __syncthreads();

    // ---------------- loop-invariant B-operands (native WMMA B layout) ----------------
    // B-matrix slice s of a 16-col tile: reg v, lane l holds W[k = 4s + v + 2*lh][col = ln]
    v2f Wg[3][8];   // w_hh^T slices for this wave's r/z/n tiles (g = i*32 + wv*16 + ln)
    v2f Wi[3][2];   // w_ih^T slices, K padded 6 -> 8 with zeros
    float biasR, biasZ, bxn, bhn;
    #pragma unroll
    for (int i = 0; i < 3; ++i) {
        const int g = i * 32 + wv * 16 + ln;
        #pragma unroll
        for (int s = 0; s < 8; ++s) {
            const int k0 = 4 * s + 2 * lh;
            Wg[i][s].x = w_hh[g * H_ + k0];
            Wg[i][s].y = w_hh[g * H_ + k0 + 1];
        }
        #pragma unroll
        for (int s = 0; s < 2; ++s) {
            const int k0 = 4 * s + 2 * lh;
            Wi[i][s].x = (k0     < I_) ? w_ih[g * I_ + k0]     : 0.0f;
            Wi[i][s].y = (k0 + 1 < I_) ? w_ih[g * I_ + k0 + 1] : 0.0f;
        }
    }
    {
        const int gr = 0 * 32 + wv * 16 + ln;
        const int gz = 1 * 32 + wv * 16 + ln;
        const int gn = 2 * 32 + wv * 16 + ln;
        biasR = b_ih[gr] + b_hh[gr];    // gx and gh merged for r/z -> fold both biases into C
        biasZ = b_ih[gz] + b_hh[gz];
        bxn   = b_ih[gn];               // n gate keeps xn / hn separate: n = tanh(xn + r*hn)
        bhn   = b_hh[gn];
    }

    // FC B-operand (wave 0 only): fc_w^T padded to 32x16 (cols >= O are zero)
    const bool fcl = (wv == 0) && (ln < O_);
    v2f Wfc[8];
    #pragma unroll
    for (int s = 0; s < 8; ++s) {
        const int k0 = 4 * s + 2 * lh;
        Wfc[s].x = fcl ? fc_w[ln * H_ + k0]     : 0.0f;
        Wfc[s].y = fcl ? fc_w[ln * H_ + k0 + 1] : 0.0f;
    }
    const float fcb = fcl ? fc_b[ln] : 0.0f;

    // lengths for the 8 D-layout rows this lane owns (M = v + 8*lh)
    int len8[8];
    #pragma unroll
    for (int v = 0; v < 8; ++v) len8[v] = lengths[b0 + v + 8 * lh];

    const float* xrow = x + (size_t)(b0 + ln) * T_ * I_;   // A-layout row M = ln
    const int    hcol = wv * 16 + ln;                      // this wave's h column (D layout)

    // h state: hA = A-layout K-slices (shared full h); hold = this lane's D-layout h values
    v2f   hA[8];
    float hold[8];
    #pragma unroll
    for (int s = 0; s < 8; ++s) { hA[s].x = 0.0f; hA[s].y = 0.0f; }
    #pragma unroll
    for (int v = 0; v < 8; ++v) hold[v] = 0.0f;

    #pragma unroll 1
    for (int t = 0; t <= T_; ++t) {
        // -------- FC of h~_{t-1} -> y_{t-1} (wave 0); masked rows emit fc_b --------
        if (wv == 0) {
            v8f accF;
            #pragma unroll
            for (int v = 0; v < 8; ++v) accF[v] = fcb;
            #pragma unroll
            for (int s = 0; s < 8; ++s) accF = wmma4(hA[s], Wfc[s], accF);
            if (t > 0 && ln < O_) {
                #pragma unroll
                for (int v = 0; v < 8; ++v) {
                    const float val = ((t - 1) < len8[v]) ? accF[v] : fcb;
                    out[((size_t)(b0 + v + 8 * lh) * T_ + (t - 1)) * O_ + ln] = val;
                }
            }
        }
        if (t == T_) break;

        // -------- load x_t in A layout (K padded 6 -> 8) --------
        const float* xp = xrow + t * I_;
        v2f xa0, xa1;
        xa0.x = xp[2 * lh];
        xa0.y = xp[2 * lh + 1];
        xa1.x = lh ? 0.0f : xp[4];
        xa1.y = lh ? 0.0f : xp[5];
        if (t + 1 < T_) __builtin_prefetch(xp + I_, 0, 1);   // global_prefetch next step

        // -------- gate pre-activations: C init with biases, accumulate gx then gh --------
        v8f accR, accZ, accXN, accHN;
        #pragma unroll
        for (int v = 0; v < 8; ++v) {
            accR[v] = biasR; accZ[v] = biasZ; accXN[v] = bxn; accHN[v] = bhn;
        }
        accR  = wmma4(xa0, Wi[0][0], accR);  accR  = wmma4(xa1, Wi[0][1], accR);
        accZ  = wmma4(xa0, Wi[1][0], accZ);  accZ  = wmma4(xa1, Wi[1][1], accZ);
        accXN = wmma4(xa0, Wi[2][0], accXN); accXN = wmma4(xa1, Wi[2][1], accXN);
        #pragma unroll
        for (int s = 0; s < 8; ++s) {
            accR  = wmma4(hA[s], Wg[0][s], accR);
            accZ  = wmma4(hA[s], Wg[1][s], accZ);
            accHN = wmma4(hA[s], Wg[2][s], accHN);
        }

        // -------- gates, GRU blend, length mask; publish h~_t to LDS --------
        #pragma unroll
        for (int v = 0; v < 8; ++v) {
            const float r  = fsigmoid(accR[v]);
            const float z  = fsigmoid(accZ[v]);
            const float n  = ftanh(accXN[v] + r * accHN[v]);
            const float hn = (1.0f - z) * n + z * hold[v];
            const float hs = (t < len8[v]) ? hn : hold[v];   // freeze finished sequences
            ldsh[hcol * 16 + 8 * lh + v] = hs;               // contiguous 8-float run
            hold[v] = hs;
        }

        __syncthreads();   // h_t fully published by both waves

        // -------- reload full h_t in A layout for the next step --------
        #pragma unroll
        for (int s = 0; s < 8; ++s) {
            const int k0 = 4 * s + 2 * lh;
            hA[s].x = ldsh[k0 * 16 + ln];
            hA[s].y = ldsh[(k0 + 1) * 16 + ln];
        }

        __syncthreads();   // reads done before anyone writes h_{t+1}
    }
}

extern "C" void kernel_launch(void* const* d_in, const int* in_sizes, int n_in,
                              void* d_out, int out_size, void* d_ws, size_t ws_size,
                              hipStream_t stream) {
    const float* x       = (const float*)d_in[0];
    const int*   lengths = (const int*)  d_in[1];
    const float* w_ih    = (const float*)d_in[2];
    const float* w_hh    = (const float*)d_in[3];
    const float* b_ih    = (const float*)d_in[4];
    const float* b_hh    = (const float*)d_in[5];
    const float* fc_w    = (const float*)d_in[6];
    const float* fc_b    = (const float*)d_in[7];
    float*       out     = (float*)d_out;

    gru_fused_kernel<<<dim3(B_ / 16), dim3(64), 0, stream>>>(
        x, lengths, w_ih, w_hh, b_ih, b_hh, fc_w, fc_b, out);
}